// GATLayer_48756468744548
// MI455X (gfx1250) — compile-verified
//
#include <hip/hip_runtime.h>

#define N_NODES   8192
#define IN_F      512
#define OUT_F     64
#define ALPHA     0.2f
#define MAXNBR    1024

typedef __attribute__((ext_vector_type(2))) float v2f;
typedef __attribute__((ext_vector_type(4))) float v4f;
typedef __attribute__((ext_vector_type(8))) float v8f;

// ---------------------------------------------------------------------------
// Kernel 1: Wh = h @ W   (8192x512 @ 512x64, exact fp32 via V_WMMA_F32_16X16X4_F32)
// Block = 128 threads (4 waves). Wave w computes the 16x16 tile at
// rows [16*blockIdx.x, +16), cols [16*w, +16). K-loop over 512 in steps of 4.
// ---------------------------------------------------------------------------
__global__ __launch_bounds__(128)
void wh_wmma_kernel(const float* __restrict__ h,
                    const float* __restrict__ W,
                    float* __restrict__ Wh) {
    const int lane = threadIdx.x & 31;
    const int col0 = (threadIdx.x >> 5) * 16;      // wave -> column tile
    const int row0 = blockIdx.x * 16;
    const int mn   = lane & 15;                    // M for A, N for B
    const int kb   = (lane >> 4) * 2;              // 0 or 2 (K sub-offset)

    const float* arow = h + (size_t)(row0 + mn) * IN_F;

    v8f c = {};
    for (int k = 0; k < IN_F; k += 4) {
        // A: 16x4 fp32 tile. lane<16: (M=mn, K=k+0,k+1); lane>=16: (M=mn, K=k+2,k+3)
        v2f a = *(const v2f*)(arow + k + kb);
        // B: 4x16 fp32 tile, same half-lane K split, N striped across lanes.
        v2f b;
        b.x = W[(size_t)(k + kb)     * OUT_F + col0 + mn];
        b.y = W[(size_t)(k + kb + 1) * OUT_F + col0 + mn];
        c = __builtin_amdgcn_wmma_f32_16x16x4_f32(
                /*neg_a=*/false, a, /*neg_b=*/false, b,
                /*c_mod=*/(short)0, c, /*reuse_a=*/false, /*reuse_b=*/false);
    }

    // C/D layout: c[v] = element (M = v + 8*(lane>>4), N = lane&15)
    const int half = lane >> 4;
#pragma unroll
    for (int v = 0; v < 8; ++v)
        Wh[(size_t)(row0 + v + 8 * half) * OUT_F + col0 + mn] = c[v];
}

// ---------------------------------------------------------------------------
// Kernel 2: f1[i] = Wh[i,:].a1 ; f2[i] = Wh[i,:].a2   (tiny, one thread/row)
// ---------------------------------------------------------------------------
__global__ __launch_bounds__(256)
void f_proj_kernel(const float* __restrict__ Wh,
                   const float* __restrict__ a,
                   float* __restrict__ f1,
                   float* __restrict__ f2) {
    __shared__ float sa[2 * OUT_F];
    if (threadIdx.x < 2 * OUT_F) sa[threadIdx.x] = a[threadIdx.x];
    __syncthreads();

    const int i = blockIdx.x * 256 + threadIdx.x;
    const v4f* whr = (const v4f*)(Wh + (size_t)i * OUT_F);
    float s1 = 0.f, s2 = 0.f;
#pragma unroll
    for (int q = 0; q < OUT_F / 4; ++q) {
        v4f v = whr[q];
        s1 += v.x * sa[q * 4 + 0] + v.y * sa[q * 4 + 1]
            + v.z * sa[q * 4 + 2] + v.w * sa[q * 4 + 3];
        s2 += v.x * sa[OUT_F + q * 4 + 0] + v.y * sa[OUT_F + q * 4 + 1]
            + v.z * sa[OUT_F + q * 4 + 2] + v.w * sa[OUT_F + q * 4 + 3];
    }
    f1[i] = s1;
    f2[i] = s2;
}

// ---------------------------------------------------------------------------
// Kernel 3: per-row masked softmax + aggregation + ELU.
// One 256-thread block per row i. Streams the 32KB adj row once (nontemporal),
// collects neighbor indices + pre-softmax logits into LDS, block-reduces
// max & sum, then gathers Wh rows (L2-resident) for the weighted sum.
// ---------------------------------------------------------------------------
__global__ __launch_bounds__(256)
void attn_kernel(const float* __restrict__ adj,
                 const float* __restrict__ Wh,
                 const float* __restrict__ f1,
                 const float* __restrict__ f2,
                 float* __restrict__ out) {
    __shared__ unsigned s_idx[MAXNBR];
    __shared__ float    s_w[MAXNBR];
    __shared__ float    s_red[256];
    __shared__ unsigned s_cnt;
    __shared__ float    s_m, s_l;

    const int tid = threadIdx.x;
    const int i   = blockIdx.x;
    if (tid == 0) s_cnt = 0u;
    __syncthreads();

    const float f1i = f1[i];
    const v4f* arow = (const v4f*)(adj + (size_t)i * N_NODES);

    // ---- Phase A: scan adj row, push neighbors, track running max of e ----
    float m = -3.0e38f;
    for (int k = tid; k < N_NODES / 4; k += 256) {
        __builtin_prefetch(&arow[k + 256], 0, 0);
        v4f av = __builtin_nontemporal_load(&arow[k]);
#pragma unroll
        for (int cmp = 0; cmp < 4; ++cmp) {
            float aval = (cmp == 0) ? av.x : (cmp == 1) ? av.y : (cmp == 2) ? av.z : av.w;
            if (aval > 0.f) {
                int j = k * 4 + cmp;
                float e = f1i + f2[j];
                e = (e > 0.f) ? e : ALPHA * e;               // leaky_relu
                unsigned p = atomicAdd(&s_cnt, 1u);          // ds_add_rtn_u32
                if (p < MAXNBR) { s_idx[p] = (unsigned)j; s_w[p] = e; }
                m = fmaxf(m, e);
            }
        }
    }
    s_red[tid] = m;
    __syncthreads();
#pragma unroll
    for (int s = 128; s > 0; s >>= 1) {
        if (tid < s) s_red[tid] = fmaxf(s_red[tid], s_red[tid + s]);
        __syncthreads();
    }
    if (tid == 0) s_m = s_red[0];
    __syncthreads();

    // ---- Phase B: exp(e - m), block-reduce the partition sum ----
    const int cnt = (int)((s_cnt < MAXNBR) ? s_cnt : MAXNBR);
    const float mm = s_m;
    float l = 0.f;
    for (int n = tid; n < cnt; n += 256) {
        float w = expf(s_w[n] - mm);
        s_w[n] = w;
        l += w;
    }
    __syncthreads();   // s_red reuse
    s_red[tid] = l;
    __syncthreads();
#pragma unroll
    for (int s = 128; s > 0; s >>= 1) {
        if (tid < s) s_red[tid] += s_red[tid + s];
        __syncthreads();
    }
    if (tid == 0) s_l = s_red[0];
    __syncthreads();

    // ---- Phase C: h' = (sum_j w_j * Wh[j]) / l ; out = elu(h') ----
    // thread = (chunk 0..3, feature 0..63); Wh gathers are 256B coalesced.
    const int f     = tid & 63;
    const int chunk = tid >> 6;
    float acc = 0.f;
    for (int n = chunk; n < cnt; n += 4)
        acc += s_w[n] * Wh[(size_t)s_idx[n] * OUT_F + f];
    __syncthreads();   // s_red reuse
    s_red[tid] = acc;
    __syncthreads();
    if (chunk == 0) {
        float r = (s_red[f] + s_red[64 + f] + s_red[128 + f] + s_red[192 + f]) / s_l;
        out[(size_t)i * OUT_F + f] = (r > 0.f) ? r : expm1f(r);   // elu
    }
}

// ---------------------------------------------------------------------------
extern "C" void kernel_launch(void* const* d_in, const int* in_sizes, int n_in,
                              void* d_out, int out_size, void* d_ws, size_t ws_size,
                              hipStream_t stream) {
    const float* h   = (const float*)d_in[0];
    const float* adj = (const float*)d_in[1];
    const float* W   = (const float*)d_in[2];
    const float* a   = (const float*)d_in[3];
    float* out = (float*)d_out;

    float* Wh = (float*)d_ws;                       // 8192*64 floats = 2 MB
    float* f1 = Wh + (size_t)N_NODES * OUT_F;       // 32 KB
    float* f2 = f1 + N_NODES;                       // 32 KB

    wh_wmma_kernel<<<N_NODES / 16, 128, 0, stream>>>(h, W, Wh);
    f_proj_kernel<<<N_NODES / 256, 256, 0, stream>>>(Wh, a, f1, f2);
    attn_kernel<<<N_NODES, 256, 0, stream>>>(adj, Wh, f1, f2, out);
}